// VectorQuantizerGT_17291538334248
// MI455X (gfx1250) — compile-verified
//
#include <hip/hip_runtime.h>

typedef __attribute__((ext_vector_type(2))) float v2f;
typedef __attribute__((ext_vector_type(8))) float v8f;

#define DIMK   16384
#define NCODES 8192
#define NB     8
#define KSPLIT 4
#define KCHUNK (DIMK / KSPLIT)   // 4096 per wave

// ---------------------------------------------------------------------------
// Kernel 1: fused  d2'[b][n] = ||e_n||^2 - 2 * (x_b . e_n)
// One 16-code tile per block, K split across 4 waves, WMMA f32 16x16x4.
// E is read exactly once (e2 accumulated from the same registers as the GEMM).
// ---------------------------------------------------------------------------
__global__ __launch_bounds__(32 * KSPLIT)
void vq_dist_kernel(const float* __restrict__ x,   // [8][16384]
                    const float* __restrict__ e,   // [8192][16384]
                    float* __restrict__ d2)        // [8][8192]
{
    __shared__ float part[KSPLIT - 1][8][16];

    const int tile  = blockIdx.x;            // 0..511
    const int wave  = threadIdx.x >> 5;      // 0..3 (K chunk)
    const int lane  = threadIdx.x & 31;
    const int m     = lane & 15;             // A row / B column within tile
    const int khalf = (lane >> 4) << 1;      // 0 or 2 (K pair selected by lane half)
    const int n0    = tile * 16;
    const int kbase = wave * KCHUNK;

    // A rows 8..15 are don't-care (their D rows live in lanes 16..31, never
    // stored) -> just alias them onto rows 0..7, no masking, uniform EXEC.
    const float* xrow = x + (size_t)(m & 7) * DIMK + kbase + khalf;
    const float* erow = e + (size_t)(n0 + m) * DIMK + kbase + khalf;

    v8f   acc = {};
    float e2p = 0.0f;

    for (int kk = 0; kk < KCHUNK; kk += 32) {
        // stream-ahead hint for the E row this lane is consuming
        __builtin_prefetch((const void*)(erow + kk + 1024), 0, 1);
#pragma unroll
        for (int k = kk; k < kk + 32; k += 4) {
            v2f a  = *(const v2f*)(xrow + k);
            v2f bb = *(const v2f*)(erow + k);
            e2p = fmaf(bb.x, bb.x, e2p);
            e2p = fmaf(bb.y, bb.y, e2p);
            // D = A(16x4) * B(4x16) + C   (f32)
            acc = __builtin_amdgcn_wmma_f32_16x16x4_f32(
                /*neg_a=*/false, a, /*neg_b=*/false, bb,
                /*c_mod=*/(short)0, acc, /*reuse_a=*/false, /*reuse_b=*/false);
        }
    }

    // e2 partial for column n lives split across lanes n and n+16
    float e2w = e2p + __shfl_xor(e2p, 16);

    // valid outputs: lanes 0..15, VGPR j -> (b=j, n=n0+lane)
    float val[8];
#pragma unroll
    for (int j = 0; j < 8; ++j) val[j] = fmaf(-2.0f, acc[j], e2w);

    if (wave != 0 && lane < 16) {
#pragma unroll
        for (int j = 0; j < 8; ++j) part[wave - 1][j][lane] = val[j];
    }
    __syncthreads();
    if (wave == 0 && lane < 16) {
#pragma unroll
        for (int j = 0; j < 8; ++j) {
            float s = val[j];
#pragma unroll
            for (int w = 0; w < KSPLIT - 1; ++w) s += part[w][j][lane];
            d2[(size_t)j * NCODES + n0 + lane] = s;
        }
    }
}

// ---------------------------------------------------------------------------
// Kernel 2: argmin over 8192 codes for one batch row per block.
// First-minimum tie-break to match jnp.argmin.
// ---------------------------------------------------------------------------
__global__ __launch_bounds__(256)
void vq_argmin_kernel(const float* __restrict__ d2, int* __restrict__ idx_out,
                      float* __restrict__ out_idx_f)
{
    __shared__ float sd[256];
    __shared__ int   si[256];
    const int b = blockIdx.x;
    const float* row = d2 + (size_t)b * NCODES;

    float best = INFINITY; int bi = NCODES;
    for (int n = threadIdx.x; n < NCODES; n += 256) {
        float v = row[n];
        if (v < best) { best = v; bi = n; }   // n strictly increasing per thread
    }
    sd[threadIdx.x] = best; si[threadIdx.x] = bi;
    __syncthreads();
    for (int s = 128; s > 0; s >>= 1) {
        if (threadIdx.x < (unsigned)s) {
            float ov = sd[threadIdx.x + s]; int oi = si[threadIdx.x + s];
            if (ov < sd[threadIdx.x] ||
                (ov == sd[threadIdx.x] && oi < si[threadIdx.x])) {
                sd[threadIdx.x] = ov; si[threadIdx.x] = oi;
            }
        }
        __syncthreads();
    }
    if (threadIdx.x == 0) { idx_out[b] = si[0]; out_idx_f[b] = (float)si[0]; }
}

// ---------------------------------------------------------------------------
// Kernel 3: gather quantized = E[idx] and per-block partial sums of (q - x)^2
// ---------------------------------------------------------------------------
__global__ __launch_bounds__(256)
void vq_quant_kernel(const float* __restrict__ x, const float* __restrict__ e,
                     const int* __restrict__ idx, float* __restrict__ q_out,
                     float* __restrict__ partials)
{
    __shared__ float sred[256];
    const int gid = blockIdx.x * 256 + threadIdx.x;    // 0..131071
    const int b   = gid >> 14;
    const int k   = gid & (DIMK - 1);
    float q  = e[(size_t)idx[b] * DIMK + k];
    float xv = x[gid];
    q_out[gid] = q;
    float d = q - xv;
    sred[threadIdx.x] = d * d;
    __syncthreads();
    for (int s = 128; s > 0; s >>= 1) {
        if (threadIdx.x < (unsigned)s) sred[threadIdx.x] += sred[threadIdx.x + s];
        __syncthreads();
    }
    if (threadIdx.x == 0) partials[blockIdx.x] = sred[0];
}

// ---------------------------------------------------------------------------
// Kernel 4: loss = 1.25 * sum(partials) / 131072   (0.25*commit + embed)
// ---------------------------------------------------------------------------
__global__ __launch_bounds__(256)
void vq_loss_kernel(const float* __restrict__ partials, float* __restrict__ loss_out)
{
    __shared__ float sred[256];
    sred[threadIdx.x] = partials[threadIdx.x] + partials[threadIdx.x + 256];
    __syncthreads();
    for (int s = 128; s > 0; s >>= 1) {
        if (threadIdx.x < (unsigned)s) sred[threadIdx.x] += sred[threadIdx.x + s];
        __syncthreads();
    }
    if (threadIdx.x == 0)
        loss_out[0] = 1.25f * sred[0] / (float)(NB * DIMK);
}

// ---------------------------------------------------------------------------
extern "C" void kernel_launch(void* const* d_in, const int* in_sizes, int n_in,
                              void* d_out, int out_size, void* d_ws, size_t ws_size,
                              hipStream_t stream)
{
    const float* x = (const float*)d_in[0];   // (8,256,8,8) -> (8,16384)
    const float* e = (const float*)d_in[1];   // (8192,16384)
    float* out = (float*)d_out;               // [quantized 131072][loss 1][idx 8]

    // workspace layout
    float* d2       = (float*)d_ws;                                  // 8*8192 f32
    float* partials = (float*)((char*)d_ws + (size_t)NB * NCODES * 4); // 512 f32
    int*   idx      = (int*)((char*)d_ws + (size_t)NB * NCODES * 4 + 512 * 4);

    float* q_out     = out;
    float* loss_out  = out + (size_t)NB * DIMK;
    float* idx_out_f = out + (size_t)NB * DIMK + 1;

    vq_dist_kernel<<<NCODES / 16, 32 * KSPLIT, 0, stream>>>(x, e, d2);
    vq_argmin_kernel<<<NB, 256, 0, stream>>>(d2, idx, idx_out_f);
    vq_quant_kernel<<<(NB * DIMK) / 256, 256, 0, stream>>>(x, e, idx, q_out, partials);
    vq_loss_kernel<<<1, 256, 0, stream>>>(partials, loss_out);
}